// MM_GNN_8289286881713
// MI455X (gfx1250) — compile-verified
//
#include <hip/hip_runtime.h>
#include <hip/hip_bf16.h>
#include <math.h>

// ---------------------------------------------------------------------------
// GAT x5 forward for MI455X (gfx1250, wave32).
// GEMMs use V_WMMA_F32_16X16X4_F32 with a zero-padded weight matrix so the
// inner loop has ZERO guards: straight global_load_b64/b32 + 2x wmma per
// K-step (previous version emitted exec-masked branchy loads). Edge
// gather/scatter remains thread-per-(edge,feature) for coalesced HBM traffic
// (that phase is the roofline bottleneck, ~2 GB on layer 1 @ 23.3 TB/s).
// ---------------------------------------------------------------------------

typedef float v2f __attribute__((ext_vector_type(2)));
typedef float v8f __attribute__((ext_vector_type(8)));

#define N_NODES 30000
#define NE      480000
#define ETOT    (NE + N_NODES)

static __device__ __forceinline__ void atomicMaxF(float* addr, float val) {
  unsigned* ua = (unsigned*)addr;
  unsigned old = *ua;
  while (__uint_as_float(old) < val) {
    unsigned assumed = old;
    old = atomicCAS(ua, assumed, __float_as_uint(val));
    if (old == assumed) break;
  }
}

__global__ void k_fill(float* p, float v, int n) {
  int i = blockIdx.x * blockDim.x + threadIdx.x;
  if (i < n) p[i] = v;
}

__global__ void k_ew_sum(const float* __restrict__ ew, float* sum, int n) {
  __shared__ float s[256];
  float v = 0.f;
  for (int i = blockIdx.x * blockDim.x + threadIdx.x; i < n; i += gridDim.x * blockDim.x)
    v += ew[i];
  s[threadIdx.x] = v; __syncthreads();
  for (int o = 128; o > 0; o >>= 1) {
    if (threadIdx.x < o) s[threadIdx.x] += s[threadIdx.x + o];
    __syncthreads();
  }
  if (threadIdx.x == 0) atomicAdd(sum, s[0]);
}

__global__ void k_mean_fin(float* scal, float inv_n) {
  if (blockIdx.x == 0 && threadIdx.x == 0) scal[1] = scal[0] * inv_n;
}

// wedot[h] = sum_c We[h*C+c] * a_e[h*C+c]   (edge-attention collapses to ea*wedot)
__global__ void k_wedot(const float* __restrict__ We, const float* __restrict__ a_e,
                        float* wedot, int H, int C) {
  int h = threadIdx.x;
  if (h < H) {
    float s = 0.f;
    for (int c = 0; c < C; ++c) s += We[h * C + c] * a_e[h * C + c];
    wedot[h] = s;
  }
}

// Zero-pad W[K,Ncol] -> Wp[Kpad,Npad] so the GEMM inner loop needs no guards.
__global__ void k_padW(const float* __restrict__ W, float* __restrict__ Wp,
                       int K, int Ncol, int Kpad, int Npad) {
  int i = blockIdx.x * blockDim.x + threadIdx.x;
  if (i >= Kpad * Npad) return;
  int k = i / Npad, n = i - k * Npad;
  Wp[i] = (k < K && n < Ncol) ? W[k * Ncol + n] : 0.f;
}

// D[M,Ncol] = A[M,K] @ Wp[Kpad,Npad] (zero-padded). One wave computes a
// 16x32 output slab (two 16x16 WMMA tiles) so each A load feeds 2 WMMAs.
// Inner loop: 1x b64 + 4x b32 loads + 2x v_wmma, no branches, EXEC all-1s.
__global__ void k_gemm_wmma(const float* __restrict__ A, const float* __restrict__ Bp,
                            float* __restrict__ D, int M, int K, int Kpad,
                            int Ncol, int Npad) {
  int lane = threadIdx.x & 31;
  int wave = (blockIdx.x * blockDim.x + threadIdx.x) >> 5;
  int tilesN2 = Npad >> 5;                      // 32-wide column slabs
  int tm = wave / tilesN2;
  int tn = wave - tm * tilesN2;
  if (tm * 16 >= M) return;                     // wave-uniform branch
  int l16 = lane & 15;
  int hi  = lane >> 4;                          // 0 -> K=0,1 ; 1 -> K=2,3
  // A row pointer: K is even for every layer, so +hi*2 keeps 8B alignment.
  const float* Arow = A + (long)(tm * 16 + l16) * K + hi * 2;
  const float* Bc   = Bp + (long)(hi * 2) * Npad + tn * 32 + l16;
  v8f acc0 = {}, acc1 = {};
  for (int k0 = 0; k0 < Kpad; k0 += 4) {
    v2f a = *(const v2f*)(Arow + k0);           // global_load_b64
    const float* b = Bc + (long)k0 * Npad;
    __builtin_prefetch(b + 8L * Npad, 0, 1);    // global_prefetch_b8 (speculative)
    v2f b0, b1;
    b0.x = b[0];
    b0.y = b[Npad];
    b1.x = b[16];
    b1.y = b[Npad + 16];
    acc0 = __builtin_amdgcn_wmma_f32_16x16x4_f32(false, a, false, b0,
                                                 (short)0, acc0, false, false);
    acc1 = __builtin_amdgcn_wmma_f32_16x16x4_f32(false, a, false, b1,
                                                 (short)0, acc1, false, false);
  }
  // C/D layout: VGPR g holds row tm*16 + hi*8 + g, column = lane%16 of tile.
  int mbase = tm * 16 + hi * 8;
  int col0  = tn * 32 + l16;
  if (col0 < Ncol) {
    for (int g = 0; g < 8; ++g)
      D[(long)(mbase + g) * Ncol + col0] = acc0[g];
  }
  if (col0 + 16 < Ncol) {
    for (int g = 0; g < 8; ++g)
      D[(long)(mbase + g) * Ncol + col0 + 16] = acc1[g];
  }
}

// al_src[n,h] = sum_c h[n,h,c]*a_s[h,c]; al_dst likewise.
__global__ void k_node_alpha(const float* __restrict__ hbuf, const float* __restrict__ a_s,
                             const float* __restrict__ a_d, float* __restrict__ als,
                             float* __restrict__ ald, int n, int H, int C) {
  int idx = blockIdx.x * blockDim.x + threadIdx.x;
  if (idx >= n * H) return;
  int node = idx / H, h = idx - node * H;
  const float* hp = hbuf + (long)node * H * C + h * C;
  float s = 0.f, d = 0.f;
  for (int c = 0; c < C; ++c) {
    float v = hp[c];
    s += v * a_s[h * C + c];
    d += v * a_d[h * C + c];
  }
  als[idx] = s;
  ald[idx] = d;
}

// logit = leaky_relu(al_src[src] + al_dst[dst] + ea*wedot); segment max into m.
__global__ void k_edge_logit(const int* __restrict__ ei, const float* __restrict__ ew,
                             const float* __restrict__ scal, const float* __restrict__ wedot,
                             const float* __restrict__ als, const float* __restrict__ ald,
                             float* __restrict__ exb, float* __restrict__ m, int H) {
  int idx = blockIdx.x * blockDim.x + threadIdx.x;
  if (idx >= ETOT * H) return;
  int e = idx / H, h = idx - e * H;
  int s, d; float w;
  if (e < NE) { s = ei[e]; d = ei[NE + e]; w = ew[e]; }
  else        { s = d = e - NE;            w = scal[1]; }   // self loop, mean weight
  float lg = als[s * H + h] + ald[d * H + h] + w * wedot[h];
  lg = lg > 0.f ? lg : 0.2f * lg;
  exb[idx] = lg;
  atomicMaxF(&m[d * H + h], lg);
}

// ex = exp(logit - m[dst]); segment sum into den.
__global__ void k_edge_exp(const int* __restrict__ ei, float* __restrict__ exb,
                           const float* __restrict__ m, float* __restrict__ den, int H) {
  int idx = blockIdx.x * blockDim.x + threadIdx.x;
  if (idx >= ETOT * H) return;
  int e = idx / H, h = idx - e * H;
  int d = (e < NE) ? ei[NE + e] : (e - NE);
  float v = __expf(exb[idx] - m[d * H + h]);
  exb[idx] = v;
  atomicAdd(&den[d * H + h], v);
}

// out[dst,f] += h[src,f] * alpha[e,h]; thread per (edge,feature) for coalescing.
__global__ void k_scatter(const int* __restrict__ ei, const float* __restrict__ hbuf,
                          const float* __restrict__ exb, const float* __restrict__ den,
                          float* __restrict__ out, int H, int C) {
  int HC = H * C;
  long total = (long)ETOT * HC;
  long idx = (long)blockIdx.x * blockDim.x + threadIdx.x;
  if (idx >= total) return;
  int e = (int)(idx / HC);
  int f = (int)(idx - (long)e * HC);
  int h = f / C;
  int s, d;
  if (e < NE) { s = ei[e]; d = ei[NE + e]; }
  else        { s = d = e - NE; }
  float alpha = exb[(long)e * H + h] / fmaxf(den[d * H + h], 1e-16f);
  atomicAdd(&out[(long)d * HC + f], hbuf[(long)s * HC + f] * alpha);
}

// One block per column: biased mean/var -> mu, rsqrt(var+eps).
__global__ void k_bn_stats(const float* __restrict__ x, float* mu, float* rsig,
                           int n, int D) {
  int d = blockIdx.x;
  float s = 0.f, q = 0.f;
  for (int r = threadIdx.x; r < n; r += blockDim.x) {
    float v = x[(long)r * D + d];
    s += v; q += v * v;
  }
  __shared__ float ss[256], sq[256];
  ss[threadIdx.x] = s; sq[threadIdx.x] = q; __syncthreads();
  for (int o = 128; o > 0; o >>= 1) {
    if (threadIdx.x < o) { ss[threadIdx.x] += ss[threadIdx.x + o]; sq[threadIdx.x] += sq[threadIdx.x + o]; }
    __syncthreads();
  }
  if (threadIdx.x == 0) {
    float m0 = ss[0] / n;
    float var = sq[0] / n - m0 * m0;
    mu[d] = m0;
    rsig[d] = rsqrtf(var + 1e-5f);
  }
}

__global__ void k_bn_apply_relu(float* __restrict__ x, const float* __restrict__ mu,
                                const float* __restrict__ rsig, const float* __restrict__ g,
                                const float* __restrict__ be, int total, int D) {
  int i = blockIdx.x * blockDim.x + threadIdx.x;
  if (i >= total) return;
  int d = i % D;
  float v = (x[i] - mu[d]) * rsig[d] * g[d] + be[d];
  x[i] = fmaxf(v, 0.f);
}

__global__ void k_bias_relu(float* __restrict__ x, const float* __restrict__ b,
                            int total, int D) {
  int i = blockIdx.x * blockDim.x + threadIdx.x;
  if (i >= total) return;
  x[i] = fmaxf(x[i] + b[i % D], 0.f);
}

__global__ void k_bias_out(const float* __restrict__ x, const float* __restrict__ b,
                           float* __restrict__ out, int n) {
  int i = blockIdx.x * blockDim.x + threadIdx.x;
  if (i < n) out[i] = x[i] + b[0];
}

static inline int cdiv(long a, long b) { return (int)((a + b - 1) / b); }

extern "C" void kernel_launch(void* const* d_in, const int* in_sizes, int n_in,
                              void* d_out, int out_size, void* d_ws, size_t ws_size,
                              hipStream_t stream) {
  const float* x  = (const float*)d_in[0];
  const int*   ei = (const int*)d_in[1];
  const float* ew = (const float*)d_in[2];
  const float *W[5], *as_[5], *ad_[5], *We[5], *ae[5], *bb[5];
  for (int l = 0; l < 5; ++l) {
    W[l]   = (const float*)d_in[3 + 6 * l + 0];
    as_[l] = (const float*)d_in[3 + 6 * l + 1];
    ad_[l] = (const float*)d_in[3 + 6 * l + 2];
    We[l]  = (const float*)d_in[3 + 6 * l + 3];
    ae[l]  = (const float*)d_in[3 + 6 * l + 4];
    bb[l]  = (const float*)d_in[3 + 6 * l + 5];
  }
  const float* g[3]  = {(const float*)d_in[33], (const float*)d_in[35], (const float*)d_in[37]};
  const float* be[3] = {(const float*)d_in[34], (const float*)d_in[36], (const float*)d_in[38]};

  // Workspace layout (floats); ~164 MB total.
  float* ws   = (float*)d_ws;
  float* O1   = ws;                                 // N*500 (out of layers 1,3,5)
  float* O2   = O1 + (size_t)N_NODES * 500;         // N*250 (out of layers 2,4)
  float* Hb   = O2 + (size_t)N_NODES * 250;         // N*500 (pre-aggregation features)
  float* als  = Hb + (size_t)N_NODES * 500;         // N*5
  float* ald  = als + (size_t)N_NODES * 5;
  float* mbuf = ald + (size_t)N_NODES * 5;
  float* den  = mbuf + (size_t)N_NODES * 5;
  float* exb  = den + (size_t)N_NODES * 5;          // ETOT*5
  float* mu   = exb + (size_t)ETOT * 5;
  float* rsig = mu + 512;
  float* scal = rsig + 512;                         // [0]=sum(ew), [1]=mean(ew)
  float* wedot = scal + 8;
  float* Wpad  = wedot + 8;                         // up to 500*256 floats

  // mean edge weight (self-loop fill value)
  k_fill<<<1, 256, 0, stream>>>(scal, 0.f, 8);
  k_ew_sum<<<512, 256, 0, stream>>>(ew, scal, NE);
  k_mean_fin<<<1, 32, 0, stream>>>(scal, 1.f / (float)NE);

  const int cin[5]   = {128, 500, 250, 128, 32};
  const int coutA[5] = {100, 50, 32, 8, 1};
  const int headA[5] = {5, 5, 4, 4, 1};
  float* outs[5] = {O1, O2, O1, O2, O1};

  const float* in = x;
  for (int l = 0; l < 5; ++l) {
    int H = headA[l], C = coutA[l], HC = H * C, K = cin[l];
    int Kpad = (K + 3) & ~3;
    int Npad = (HC + 31) & ~31;
    float* out = outs[l];

    // pad weights, then h = in @ W via guard-free WMMA loop
    k_padW<<<cdiv((long)Kpad * Npad, 256), 256, 0, stream>>>(W[l], Wpad, K, HC, Kpad, Npad);
    int tilesM = cdiv(N_NODES, 16), tilesN2 = Npad >> 5;
    long waves = (long)tilesM * tilesN2;
    k_gemm_wmma<<<cdiv(waves * 32, 256), 256, 0, stream>>>(
        in, Wpad, Hb, N_NODES, K, Kpad, HC, Npad);

    k_wedot<<<1, 32, 0, stream>>>(We[l], ae[l], wedot, H, C);
    k_node_alpha<<<cdiv((long)N_NODES * H, 256), 256, 0, stream>>>(
        Hb, as_[l], ad_[l], als, ald, N_NODES, H, C);

    k_fill<<<cdiv((long)N_NODES * H, 256), 256, 0, stream>>>(mbuf, -INFINITY, N_NODES * H);
    k_fill<<<cdiv((long)N_NODES * H, 256), 256, 0, stream>>>(den, 0.f, N_NODES * H);
    k_fill<<<cdiv((long)N_NODES * HC, 256), 256, 0, stream>>>(out, 0.f, N_NODES * HC);

    k_edge_logit<<<cdiv((long)ETOT * H, 256), 256, 0, stream>>>(
        ei, ew, scal, wedot, als, ald, exb, mbuf, H);
    k_edge_exp<<<cdiv((long)ETOT * H, 256), 256, 0, stream>>>(ei, exb, mbuf, den, H);
    k_scatter<<<cdiv((long)ETOT * HC, 256), 256, 0, stream>>>(ei, Hb, exb, den, out, H, C);

    if (l < 3) {
      // GAT bias cancels inside BatchNorm (mean shift) -> skip it.
      k_bn_stats<<<HC, 256, 0, stream>>>(out, mu, rsig, N_NODES, HC);
      k_bn_apply_relu<<<cdiv((long)N_NODES * HC, 256), 256, 0, stream>>>(
          out, mu, rsig, g[l], be[l], N_NODES * HC, HC);
    } else if (l == 3) {
      k_bias_relu<<<cdiv((long)N_NODES * HC, 256), 256, 0, stream>>>(
          out, bb[l], N_NODES * HC, HC);
    } else {
      k_bias_out<<<cdiv(N_NODES, 256), 256, 0, stream>>>(out, bb[l], (float*)d_out, N_NODES);
    }
    in = out;
  }
}